// IntensitiyTransform_57097295233086
// MI455X (gfx1250) — compile-verified
//
#include <hip/hip_runtime.h>

// IntensityTransform: out[b,h,w,c] = transforms[b, round(255*(0.5*img+0.5)), c]
// images:     [32, 512, 512, 3] f32  (~100.7 MB)
// transforms: [32, 256, 3]      f32  (3 KB per batch)
// Purely HBM-bound: ~201 MB traffic -> ~8.6 us floor at 23.3 TB/s.
// Strategy: b128 vector loads/stores with non-temporal hints; per-batch LUT
// staged into LDS via the gfx1250 async global->LDS path; ds_load gather.

typedef float v4f __attribute__((ext_vector_type(4)));
typedef int   v4i __attribute__((ext_vector_type(4)));

// gfx1250 async-copy builtin takes typed int4 pointers:
//   param0: addrspace(1) int4*  (global source)
//   param1: addrspace(3) pointer (LDS destination)
typedef __attribute__((address_space(1))) v4i* global_v4i_ptr;
typedef __attribute__((address_space(3))) v4i* lds_v4i_ptr;

#define LUT_FLOATS 768   // 256 intensities * 3 channels
#define BLOCK 256        // 8 wave32s per block

__global__ __launch_bounds__(BLOCK) void intensity_lut_kernel(
    const float* __restrict__ images,
    const float* __restrict__ transforms,
    float* __restrict__ out,
    int vec4_per_image)
{
    __shared__ float lut[LUT_FLOATS];

    const int b = blockIdx.y;
    const float* __restrict__ lut_g = transforms + (size_t)b * LUT_FLOATS;

    // ---- Stage this batch's 3 KB LUT into LDS ----
#if __has_builtin(__builtin_amdgcn_global_load_async_to_lds_b128)
    // 768 floats = 192 x b128; lanes 0..191 issue one async copy each.
    if (threadIdx.x < LUT_FLOATS / 4) {
        __builtin_amdgcn_global_load_async_to_lds_b128(
            (global_v4i_ptr)(lut_g + threadIdx.x * 4),
            (lds_v4i_ptr)(&lut[threadIdx.x * 4]),
            /*offset=*/0, /*cpol=*/0);
    }
  #if __has_builtin(__builtin_amdgcn_s_wait_asynccnt)
    __builtin_amdgcn_s_wait_asynccnt(0);
  #else
    asm volatile("s_wait_asynccnt 0" ::: "memory");
  #endif
#else
    // Fallback: plain global load + ds store.
    for (int i = threadIdx.x; i < LUT_FLOATS / 4; i += BLOCK) {
        ((v4f*)lut)[i] = ((const v4f*)lut_g)[i];
    }
#endif
    __syncthreads();

    const size_t img_base = (size_t)b * (size_t)vec4_per_image * 4u;
    const v4f* __restrict__ in4 = (const v4f*)(images + img_base);
    v4f* __restrict__ out4      = (v4f*)(out + img_base);

    // Grid-stride over the image in float4 units. Non-temporal: the 200 MB
    // stream would otherwise evict everything useful from L2.
    for (int i = blockIdx.x * BLOCK + (int)threadIdx.x; i < vec4_per_image;
         i += gridDim.x * BLOCK) {
        v4f v = __builtin_nontemporal_load(&in4[i]);
        const int c0 = i % 3;  // (4*i) % 3 == i % 3
        v4f o;
#pragma unroll
        for (int k = 0; k < 4; ++k) {
            // idx = round_half_even(127.5*x + 127.5), matching jnp.round (RNE)
            int idx = (int)rintf(fmaf(v[k], 127.5f, 127.5f));
            idx = idx < 0 ? 0 : (idx > 255 ? 255 : idx);
            int c = c0 + k;          // 0..5
            if (c >= 3) c -= 3;      // 0..2
            o[k] = lut[idx * 3 + c]; // ds_load_b32 gather from LDS
        }
        __builtin_nontemporal_store(o, &out4[i]);
    }
}

extern "C" void kernel_launch(void* const* d_in, const int* in_sizes, int n_in,
                              void* d_out, int out_size, void* d_ws, size_t ws_size,
                              hipStream_t stream) {
    const float* images     = (const float*)d_in[0];
    const float* transforms = (const float*)d_in[1];
    float* out              = (float*)d_out;

    const int n_img = in_sizes[0];              // B*H*W*C floats
    const int n_tr  = in_sizes[1];              // B*256*3 floats
    const int B     = n_tr / LUT_FLOATS;        // 32
    const int per_img_f  = n_img / B;           // 786432
    const int per_img_v4 = per_img_f / 4;       // 196608 float4s

    // ~8 float4 iterations per thread: 196608 / (256*8) = 96 blocks per image.
    int blocks_x = (per_img_v4 + BLOCK * 8 - 1) / (BLOCK * 8);
    if (blocks_x < 1) blocks_x = 1;

    dim3 grid(blocks_x, B);
    dim3 block(BLOCK);
    intensity_lut_kernel<<<grid, block, 0, stream>>>(images, transforms, out,
                                                     per_img_v4);
}